// _TemplateProposalLayer_3908420239688
// MI455X (gfx1250) — compile-verified
//
#include <hip/hip_runtime.h>
#include <stdint.h>

// Problem constants (from the reference setup)
#define BB   8          // batch
#define KK   2048       // gt boxes per image (== candidate rois per image)
#define RPI  256        // ROIS_PER_IMAGE
#define FGP  64         // FG_PER_IMAGE
#define BIGC 1.0e6f
#define NTHR 256

typedef uint32_t u32x4 __attribute__((ext_vector_type(4)));
typedef uint32_t u32x8 __attribute__((ext_vector_type(8)));

// ---------------------------------------------------------------------------
// Kernel 1: build candidate boxes.
//   cand layout: [it][b][k][4] floats (x1,y1,x2,y2); it=0 jittered, it=1 raw.
// ---------------------------------------------------------------------------
__global__ void tpl_prep_kernel(const float* __restrict__ gt,
                                const float* __restrict__ noise,
                                const int* __restrict__ pW,
                                const int* __restrict__ pH,
                                float* __restrict__ cand) {
    int i = blockIdx.x * blockDim.x + threadIdx.x;   // over B*K
    if (i >= BB * KK) return;
    const float* g = gt + (size_t)i * 6;
    float x1 = g[0], y1 = g[1], x2 = g[2], y2 = g[3];

    // raw boxes -> iteration 1
    float* c1 = cand + (size_t)(BB * KK * 4) + (size_t)i * 4;
    c1[0] = x1; c1[1] = y1; c1[2] = x2; c1[3] = y2;

    // jittered boxes -> iteration 0
    const float* nz = noise + (size_t)i * 4;
    float wmax = (float)pW[0] - 1.0f;
    float hmax = (float)pH[0] - 1.0f;
    float x = (x1 + x2) * 0.5f;
    float y = (y1 + y2) * 0.5f;
    float w = x2 - x1;
    float h = y2 - y1;
    float nx = nz[0] * 0.3f * w + x;
    float ny = nz[1] * 0.3f * h + y;
    float nw = expf(nz[2] * 0.5f) * w;
    float nh = expf(nz[3] * 0.5f) * h;
    float jx1 = fminf(fmaxf(nx - nw * 0.5f, 0.0f), wmax);
    float jy1 = fminf(fmaxf(ny - nh * 0.5f, 0.0f), hmax);
    float jx2 = fminf(fmaxf(nx + nw * 0.5f, 0.0f), wmax);
    float jy2 = fminf(fmaxf(ny + nh * 0.5f, 0.0f), hmax);
    float* c0 = cand + (size_t)i * 4;
    c0[0] = jx1; c0[1] = jy1; c0[2] = jx2; c0[3] = jy2;
}

// ---------------------------------------------------------------------------
// Kernel 2: per-ROI max IoU + argmax vs 2048 GT boxes.
// GT tile staged global->LDS with ONE Tensor Data Mover descriptor:
//   2D tensor, rows of 6 dwords, tile = 4 dwords x 2048 rows, compacted
//   contiguously into LDS (TENSORcnt path).
// grid = (KK/NTHR, BB, 2), block = NTHR.
// ---------------------------------------------------------------------------
__global__ __launch_bounds__(NTHR)
void tpl_iou_kernel(const float* __restrict__ gt,
                    const float* __restrict__ cand,
                    float* __restrict__ maxov,
                    int* __restrict__ amax) {
    __shared__ __align__(16) float sg[KK * 4];   // 32 KB: gt (x1,y1,x2,y2) per k

    const int b  = blockIdx.y;
    const int it = blockIdx.z;
    const float* gb = gt + (size_t)b * KK * 6;

    if (threadIdx.x < 32) {                       // one wave issues the TDM op
        uint64_t gaddr = (uint64_t)(uintptr_t)gb;             // 57-bit VA
        uint32_t laddr = (uint32_t)(uintptr_t)(&sg[0]);       // LDS byte offset
        // D# group 0: count=1 | lds_addr | global_addr[56:0] | type=2
        u32x4 g0;
        g0.x = 1u;
        g0.y = laddr;
        g0.z = (uint32_t)gaddr;
        g0.w = (uint32_t)((gaddr >> 32) & 0x01FFFFFFu) | (2u << 30);
        // D# group 1: data_size=4B; tensor_dim0=4, tensor_dim1=2048;
        //             tile_dim0=4, tile_dim1=2048; tensor_dim0_stride=6
        u32x8 g1;
        g1.s0 = 0x00020000u;          // workgroup_mask=0, data_size=2 (4B)
        g1.s1 = (4u << 16);           // tensor_dim0[15:0] in bits 63:48
        g1.s2 = (2048u << 16);        // tensor_dim0 hi=0; tensor_dim1[15:0]
        g1.s3 = (4u << 16);           // tensor_dim1 hi=0; tile_dim0=4
        g1.s4 = 2048u;                // tile_dim1=2048, tile_dim2=0 (2D)
        g1.s5 = 6u;                   // tensor_dim0_stride = 6 dwords
        g1.s6 = 0u;                   // stride hi, tensor_dim1_stride lo
        g1.s7 = 0u;                   // tensor_dim1_stride hi
        u32x4 gz = {0u, 0u, 0u, 0u};  // groups 2/3 unused (2D tile)
        asm volatile("tensor_load_to_lds %0, %1, %2, %3"
                     :: "s"(g0), "s"(g1), "s"(gz), "s"(gz) : "memory");
        __builtin_amdgcn_s_wait_tensorcnt(0);     // DMA complete for this wave
    }
    __syncthreads();                              // publish LDS tile to all waves

    const int n = blockIdx.x * NTHR + threadIdx.x;    // roi index within image
    const float* c = cand + (size_t)(((size_t)it * BB + b) * KK + n) * 4;
    float ax1 = c[0], ay1 = c[1], ax2 = c[2], ay2 = c[3];
    float area_r = (ax2 - ax1 + 1.0f) * (ay2 - ay1 + 1.0f);

    float best = -__builtin_inff();
    int   bi   = 0;
    for (int k = 0; k < KK; ++k) {
        float4 g = *(const float4*)(&sg[k * 4]);      // ds_load_b128, broadcast
        float gw = g.z - g.x + 1.0f;
        float gh = g.w - g.y + 1.0f;
        float area_g = gw * gh;
        float ix1 = fmaxf(ax1, g.x);
        float iy1 = fmaxf(ay1, g.y);
        float ix2 = fminf(ax2, g.z);
        float iy2 = fminf(ay2, g.w);
        float iw = fmaxf(ix2 - ix1 + 1.0f, 0.0f);
        float ih = fmaxf(iy2 - iy1 + 1.0f, 0.0f);
        float inter = iw * ih;
        float iou = inter / (area_r + area_g - inter);
        if (iou > best) { best = iou; bi = k; }       // first-max wins (jnp.argmax)
    }
    size_t o = ((size_t)it * BB + b) * KK + n;
    maxov[o] = best;
    amax[o]  = bi;
}

// ---------------------------------------------------------------------------
// Kernel 3: per-(image, iteration) sampling: fg/bg counts, two stable
// argsorts (64-bit bitonic: ordered-float key << 32 | index), gather + write.
// maxov row staged global->LDS via async b128 copies (ASYNCcnt path).
// grid = (BB, 2), block = NTHR.
// ---------------------------------------------------------------------------
__device__ __forceinline__ uint32_t ordered_f32(float f) {
    uint32_t u = __float_as_uint(f);
    return (u & 0x80000000u) ? ~u : (u | 0x80000000u);   // ascending uint == ascending float
}

__device__ void bitonic_sort_2048(uint64_t* a, int tid) {
    for (unsigned k = 2; k <= KK; k <<= 1) {
        for (unsigned j = k >> 1; j > 0; j >>= 1) {
            for (unsigned t = tid; t < KK; t += NTHR) {
                unsigned x = t ^ j;
                if (x > t) {
                    uint64_t u = a[t], v = a[x];
                    bool asc = ((t & k) == 0);
                    if (asc ? (u > v) : (u < v)) { a[t] = v; a[x] = u; }
                }
            }
            __syncthreads();
        }
    }
}

__global__ __launch_bounds__(NTHR)
void tpl_sample_kernel(const float* __restrict__ gt,
                       const float* __restrict__ cand,
                       const float* __restrict__ maxov,
                       const int* __restrict__ amax,
                       float* __restrict__ out) {
    __shared__ uint64_t skey[KK];                 // 16 KB sort buffer
    __shared__ __align__(16) float sov[KK];       // 8 KB staged max-overlaps
    __shared__ int forder[RPI];
    __shared__ int border[RPI];
    __shared__ int s_nfg, s_nbg;

    const int b   = blockIdx.x;
    const int it  = blockIdx.y;
    const int tid = threadIdx.x;
    const size_t base = ((size_t)it * BB + b) * KK;

    if (tid == 0) { s_nfg = 0; s_nbg = 0; }

    // Async global -> LDS stage of this image/iteration's 2048 overlaps.
    for (int i = tid; i < KK / 4; i += NTHR) {    // 512 x b128, 2 per thread
        uint32_t lds = (uint32_t)(uintptr_t)(&sov[i * 4]);
        uint64_t ga  = (uint64_t)(uintptr_t)(maxov + base + (size_t)i * 4);
        asm volatile("global_load_async_to_lds_b128 %0, %1, off"
                     :: "v"(lds), "v"(ga) : "memory");
    }
    asm volatile("s_wait_asynccnt 0" ::: "memory");
    __syncthreads();

    // Count fg/bg, build fg sort keys.
    int nf = 0, nb = 0;
    for (int i = 0; i < KK / NTHR; ++i) {
        int k = tid + i * NTHR;
        float ov = sov[k];
        bool fg = (ov >= 0.5f);
        bool bg = ((ov < 0.5f) && (ov >= 0.0f)) || (ov < 0.0f);
        nf += fg ? 1 : 0;
        nb += bg ? 1 : 0;
        // fg key (replicates float32 rounding of fg*1e6 + ov); sort by -key asc
        float keyf = (fg ? BIGC : 0.0f) + ov;
        float sv = -keyf;
        skey[k] = ((uint64_t)ordered_f32(sv) << 32) | (uint32_t)k;
    }
    atomicAdd(&s_nfg, nf);
    atomicAdd(&s_nbg, nb);
    __syncthreads();

    bitonic_sort_2048(skey, tid);
    if (tid < RPI) forder[tid] = (int)(uint32_t)skey[tid];
    __syncthreads();

    // bg keys
    for (int i = 0; i < KK / NTHR; ++i) {
        int k = tid + i * NTHR;
        float ov = sov[k];
        bool bg = ((ov < 0.5f) && (ov >= 0.0f)) || (ov < 0.0f);
        float keyb = (bg ? BIGC : 0.0f) - ov;
        float sv = -keyb;
        skey[k] = ((uint64_t)ordered_f32(sv) << 32) | (uint32_t)k;
    }
    __syncthreads();

    bitonic_sort_2048(skey, tid);
    if (tid < RPI) border[tid] = (int)(uint32_t)skey[tid];
    __syncthreads();

    // Final selection + writeout: one thread per output slot.
    const int n_fg = s_nfg, n_bg = s_nbg;
    int fg_take = (n_bg == 0) ? RPI : ((n_fg == 0) ? 0 : min(FGP, n_fg));
    int pos = tid;                                    // 0..255
    bool is_fg = pos < fg_take;
    int mf = max(n_fg, 1), mb = max(n_bg, 1);
    int fg_idx = forder[pos % mf];                    // pos%mf <= 255 always
    int bg_idx = border[max(pos - fg_take, 0) % mb];
    int keep = is_fg ? fg_idx : bg_idx;

    int ga = amax[base + keep];
    const float* gk = gt + ((size_t)b * KK + ga) * 6;
    float lab = is_fg ? gk[4] : 0.0f;
    float trk = is_fg ? gk[5] : -1.0f;
    const float* cb = cand + (base + keep) * 4;

    const int j = it * RPI + pos;                     // 0..511 within image
    float* ro = out + ((size_t)b * (2 * RPI) + j) * 5;
    ro[0] = (float)b;
    ro[1] = cb[0]; ro[2] = cb[1]; ro[3] = cb[2]; ro[4] = cb[3];
    const size_t lab_base = (size_t)BB * 2 * RPI * 5;
    const size_t trk_base = lab_base + (size_t)BB * 2 * RPI;
    out[lab_base + (size_t)b * 2 * RPI + j] = lab;
    out[trk_base + (size_t)b * 2 * RPI + j] = trk;
}

// ---------------------------------------------------------------------------
extern "C" void kernel_launch(void* const* d_in, const int* in_sizes, int n_in,
                              void* d_out, int out_size, void* d_ws, size_t ws_size,
                              hipStream_t stream) {
    const float* gt    = (const float*)d_in[0];   // [8,2048,6]
    const float* noise = (const float*)d_in[1];   // [8,2048,4]
    const int*   pW    = (const int*)d_in[2];     // scalar 1920
    const int*   pH    = (const int*)d_in[3];     // scalar 1080
    float* out = (float*)d_out;                   // rois(20480) ++ labels(4096) ++ tracks(4096)

    // workspace: cand [2][B][K][4] f32, maxov [2][B][K] f32, amax [2][B][K] i32
    float* cand  = (float*)d_ws;
    float* maxov = cand + (size_t)2 * BB * KK * 4;
    int*   amax  = (int*)(maxov + (size_t)2 * BB * KK);

    tpl_prep_kernel<<<(BB * KK + NTHR - 1) / NTHR, NTHR, 0, stream>>>(
        gt, noise, pW, pH, cand);

    dim3 g2(KK / NTHR, BB, 2);
    tpl_iou_kernel<<<g2, NTHR, 0, stream>>>(gt, cand, maxov, amax);

    dim3 g3(BB, 2);
    tpl_sample_kernel<<<g3, NTHR, 0, stream>>>(gt, cand, maxov, amax, out);
}